// MultiHeadGraphAttention_8924942041812
// MI455X (gfx1250) — compile-verified
//
#include <hip/hip_runtime.h>

#define N_NODES 100000
#define N_EDGES 1600000
#define D 64
#define LEAKY 0.01f

typedef __attribute__((ext_vector_type(2))) float v2f;
typedef __attribute__((ext_vector_type(8))) float v8f;

// ws layout (floats):
//   s      : N_NODES*4          projected scores (src0,dst0,src1,dst1 per node)
//   rowsum : 2*N_NODES          per-(head,node) attention denominators
//   Vpad   : 16*2*16*2 = 1024   zero-padded B-matrix table for the WMMA stage
//            Vpad[c][hi][col][kk] = (col<4) ? w[h][d]*a[h][part*D+d] : 0
//            with d = 4c + 2*hi + kk, h = col>>1, part = col&1

// ---------------------------------------------------------------------------
// Kernel 1: zero output accumulator + rowsum, and build the padded V table.
// ---------------------------------------------------------------------------
__global__ void __launch_bounds__(256) gat_init_kernel(
        float* __restrict__ out, float* __restrict__ rowsum,
        float* __restrict__ vpad,
        const float* __restrict__ w, const float* __restrict__ a) {
    int idx = blockIdx.x * blockDim.x + threadIdx.x;
    if (idx < 2 * N_NODES * D) out[idx] = 0.0f;
    if (idx < 2 * N_NODES)     rowsum[idx] = 0.0f;
    if (idx < 1024) {
        const int c   = idx >> 6;          // K-chunk (0..15)
        const int hi  = (idx >> 5) & 1;    // lane-half
        const int col = (idx >> 1) & 15;   // B/D column
        const int kk  = idx & 1;           // element within float2
        const int d   = 4 * c + 2 * hi + kk;
        float v = 0.0f;
        if (col < 4) {
            const int head = col >> 1;
            const int part = col & 1;
            v = w[head * D + d] * a[head * 2 * D + part * D + d];
        }
        vpad[idx] = v;
    }
}

// ---------------------------------------------------------------------------
// Kernel 2: fused projection via fp32 WMMA.  One wave per 16-node tile:
// D(16x16) += A(16x4) x B(4x16) over 16 K-chunks (v_wmma_f32_16x16x4_f32).
// A fragment: b64 load from the node row of x; B fragment: b64 load from the
// precomputed Vpad table at an immediate offset (loop-invariant base, L1-hot).
//
// ISA layouts (CDNA5 ISA §7.12.2):
//   A 16x4 f32 : lane L<16 -> row L, VGPRs {K0,K1}; lane L+16 -> {K2,K3}
//   B  4x16    : lane n<16 -> col n, VGPRs {K0,K1}; lane n+16 -> {K2,K3}
//   D 16x16    : VGPR j, lanes 0-15 -> row j, lanes 16-31 -> row j+8
// ---------------------------------------------------------------------------
__global__ void __launch_bounds__(256) gat_score_wmma_kernel(
        const float* __restrict__ x, const float* __restrict__ vpad,
        float* __restrict__ s) {
    const int tile = blockIdx.x * 8 + (threadIdx.x >> 5);   // one wave per tile
    const int NT = N_NODES / 16;                            // 6250 exact
    if (tile >= NT) return;                                 // wave-uniform exit

    const int lane = threadIdx.x & 31;
    const int r    = lane & 15;          // A-row (loading) / B,D-column
    const int hi   = lane >> 4;          // K half-select

    const float* xrow  = x + (size_t)(tile * 16 + r) * D + 2 * hi;
    const float* vbase = vpad + hi * 32 + r * 2;

    v8f acc = {};
    #pragma unroll
    for (int c = 0; c < 16; ++c) {
        const v2f av = *(const v2f*)(xrow + c * 4);
        const v2f bv = *(const v2f*)(vbase + c * 64);
        acc = __builtin_amdgcn_wmma_f32_16x16x4_f32(
                false, av, false, bv, (short)0, acc, false, false);
    }

    if (r < 4) {
        const int rowbase = tile * 16 + 8 * hi;
        #pragma unroll
        for (int j = 0; j < 8; ++j)
            s[(rowbase + j) * 4 + r] = acc[j];
    }
}

// ---------------------------------------------------------------------------
// Kernel 3: per-edge attention + scatter-add. One wave per edge, both heads.
// Lane l handles feature dims 2l, 2l+1 (float2 gather of x[dst], 4 native
// global_atomic_add_f32 per lane). src/dst broadcast via readfirstlane.
// ---------------------------------------------------------------------------
__global__ void __launch_bounds__(256) gat_edge_kernel(
        const float* __restrict__ x, const long long* __restrict__ ei,
        const float* __restrict__ w, const float* __restrict__ s,
        float* __restrict__ rowsum, float* __restrict__ out) {
    const int wv = blockIdx.x * 8 + (threadIdx.x >> 5);
    const int lane = threadIdx.x & 31;
    if (wv >= N_EDGES) return;                               // wave-uniform

    const int srcn = __builtin_amdgcn_readfirstlane((int)ei[wv]);
    const int dstn = __builtin_amdgcn_readfirstlane((int)ei[N_EDGES + wv]);

    const float sc0 = s[srcn * 4 + 0] + s[dstn * 4 + 1];
    const float sc1 = s[srcn * 4 + 2] + s[dstn * 4 + 3];
    const float lr0 = sc0 > 0.0f ? sc0 : LEAKY * sc0;
    const float lr1 = sc1 > 0.0f ? sc1 : LEAKY * sc1;
    const float att0 = __expf(-lr0);
    const float att1 = __expf(-lr1);

    const v2f xd = ((const v2f*)(x + (size_t)dstn * D))[lane];
    const v2f w0 = ((const v2f*)w)[lane];        // head 0, dims 2l,2l+1
    const v2f w1 = ((const v2f*)w)[32 + lane];   // head 1

    float* o0 = out + (size_t)srcn * D + 2 * lane;
    float* o1 = out + (size_t)N_NODES * D + (size_t)srcn * D + 2 * lane;
    atomicAdd(o0 + 0, att0 * xd[0] * w0[0]);
    atomicAdd(o0 + 1, att0 * xd[1] * w0[1]);
    atomicAdd(o1 + 0, att1 * xd[0] * w1[0]);
    atomicAdd(o1 + 1, att1 * xd[1] * w1[1]);

    if (lane == 0) {
        atomicAdd(&rowsum[srcn], att0);
        atomicAdd(&rowsum[N_NODES + srcn], att1);
    }
}

// ---------------------------------------------------------------------------
// Kernel 4: out[h][n][d] /= rowsum[h][n]   (0/0 -> NaN matches reference)
// ---------------------------------------------------------------------------
__global__ void __launch_bounds__(256) gat_finalize_kernel(
        float* __restrict__ out, const float* __restrict__ rowsum) {
    int idx = blockIdx.x * blockDim.x + threadIdx.x;
    if (idx >= 2 * N_NODES * D) return;
    out[idx] = out[idx] / rowsum[idx >> 6];   // idx>>6 == h*N + n (D == 64)
}

// ---------------------------------------------------------------------------
extern "C" void kernel_launch(void* const* d_in, const int* in_sizes, int n_in,
                              void* d_out, int out_size, void* d_ws, size_t ws_size,
                              hipStream_t stream) {
    const float*      x  = (const float*)d_in[0];       // (N, 64) fp32
    const long long*  ei = (const long long*)d_in[1];   // (2, E) int64
    const float*      w  = (const float*)d_in[2];       // (2, 1, 64) fp32
    const float*      a  = (const float*)d_in[3];       // (2, 128, 1) fp32
    float* out = (float*)d_out;                         // (2, N, 64) fp32

    float* s      = (float*)d_ws;                       // N*4
    float* rowsum = s + (size_t)N_NODES * 4;            // 2*N
    float* vpad   = rowsum + (size_t)2 * N_NODES;       // 1024

    const int zeroBlocks  = (2 * N_NODES * D + 255) / 256;   // 50000
    const int scoreBlocks = (N_NODES / 16 + 7) / 8;          // 782 (8 waves/blk)
    const int edgeBlocks  = (N_EDGES + 7) / 8;               // 200000

    gat_init_kernel<<<zeroBlocks, 256, 0, stream>>>(out, rowsum, vpad, w, a);
    gat_score_wmma_kernel<<<scoreBlocks, 256, 0, stream>>>(x, vpad, s);
    gat_edge_kernel<<<edgeBlocks, 256, 0, stream>>>(x, ei, w, s, rowsum, out);
    gat_finalize_kernel<<<zeroBlocks, 256, 0, stream>>>(out, rowsum);
}